// PlaylistEmbedding_77421080477871
// MI455X (gfx1250) — compile-verified
//
#include <hip/hip_runtime.h>

// PlaylistEmbedding: out[b,:] = sum_n values[b,n] * w[indices[b,n],:] + bias
// B=16384 rows, NNZ=200, EMB=32. Gather-bound; w (10.4MB) is L2-resident.
// Wave32 layout: one wave per row, lane == embedding dim -> every w-row load
// is one coalesced 128B request; output store is one coalesced 128B store.
// CDNA5 paths: global_load_async_to_lds_b128/b32 (ASYNCcnt) staging of the
// idx/val stream, global_prefetch_b8 one chunk ahead of the w gathers.

#define K_NNZ 200
#define K_EMB 32
#define K_CH 8
#define K_NCH (K_NNZ / K_CH)   // 25 chunks

#ifndef __has_builtin
#define __has_builtin(x) 0
#endif

#if __has_builtin(__builtin_amdgcn_global_load_async_to_lds_b32)
#define USE_ASYNC_LDS 1
typedef __attribute__((address_space(1))) int gas_int;   // global AS
typedef __attribute__((address_space(3))) int las_int;   // LDS AS
#else
#define USE_ASYNC_LDS 0
#endif

#if __has_builtin(__builtin_amdgcn_global_load_async_to_lds_b128)
#define USE_ASYNC_B128 1
typedef int v4i __attribute__((ext_vector_type(4)));
typedef __attribute__((address_space(1))) v4i gas_v4i;
typedef __attribute__((address_space(3))) v4i las_v4i;
#else
#define USE_ASYNC_B128 0
#endif

__device__ __forceinline__ void wait_asynccnt0() {
#if __has_builtin(__builtin_amdgcn_s_wait_asynccnt)
    __builtin_amdgcn_s_wait_asynccnt(0);
#else
    asm volatile("s_wait_asynccnt 0" ::: "memory");
#endif
}

__global__ __launch_bounds__(32)
void playlist_embedding_kernel(const int* __restrict__ indices,
                               const float* __restrict__ values,
                               const float* __restrict__ w,
                               const float* __restrict__ bias,
                               float* __restrict__ out)
{
    const int b    = blockIdx.x;    // one wave32 per output row
    const int lane = threadIdx.x;   // embedding dimension 0..31

    const int*   ib = indices + (size_t)b * K_NNZ;   // 800B, 16B-aligned
    const float* vb = values  + (size_t)b * K_NNZ;   // 800B, 16B-aligned

#if USE_ASYNC_LDS
    __shared__ __attribute__((aligned(16))) int   s_idx[K_NNZ];
    __shared__ __attribute__((aligned(16))) float s_val[K_NNZ];
#if USE_ASYNC_B128
    // 800B per stream == 25 lanes x 32B: one predicated pass, 4 async b128
    // per participating lane; imm offset covers the second 16B (ISA adds
    // INST_OFFSET to both the global and the LDS address).
    if (lane < 25) {
        const int* gi = ib + lane * 8;
        const int* gv = (const int*)(vb + lane * 8);
        __builtin_amdgcn_global_load_async_to_lds_b128(
            (gas_v4i*)gi, (las_v4i*)&s_idx[lane * 8], 0, 0);
        __builtin_amdgcn_global_load_async_to_lds_b128(
            (gas_v4i*)gi, (las_v4i*)&s_idx[lane * 8], 16, 0);
        __builtin_amdgcn_global_load_async_to_lds_b128(
            (gas_v4i*)gv, (las_v4i*)&s_val[lane * 8], 0, 0);
        __builtin_amdgcn_global_load_async_to_lds_b128(
            (gas_v4i*)gv, (las_v4i*)&s_val[lane * 8], 16, 0);
    }
#else
    for (int t = lane; t < K_NNZ; t += 32) {
        __builtin_amdgcn_global_load_async_to_lds_b32(
            (gas_int*)(ib + t), (las_int*)&s_idx[t], 0, 0);
        __builtin_amdgcn_global_load_async_to_lds_b32(
            (gas_int*)(vb + t), (las_int*)&s_val[t], 0, 0);
    }
#endif
    wait_asynccnt0();   // single wave per block: no barrier needed
    const int*   ip = s_idx;   // wave-uniform LDS reads -> ds broadcast
    const float* vp = s_val;
#else
    const int*   ip = ib;      // wave-uniform global reads (scalar-friendly)
    const float* vp = vb;
#endif

    float acc = 0.0f;

    // Software pipeline: registers hold current chunk's idx/val while the
    // next chunk is fetched and its w rows are prefetched (global_prefetch_b8).
    int   ci[K_CH];
    float cv[K_CH];
#pragma unroll
    for (int j = 0; j < K_CH; ++j) { ci[j] = ip[j]; cv[j] = vp[j]; }

#pragma unroll 1
    for (int c = 0; c < K_NCH; ++c) {
        int   ni[K_CH];
        float nv[K_CH];
        const bool more = (c + 1 < K_NCH);
        if (more) {
#pragma unroll
            for (int j = 0; j < K_CH; ++j) {
                ni[j] = ip[(c + 1) * K_CH + j];
                nv[j] = vp[(c + 1) * K_CH + j];
            }
            // Pull next chunk's gathered rows toward the WGP while we FMA
            // the current chunk. Lanes cover the full 128B row.
#pragma unroll
            for (int j = 0; j < K_CH; ++j) {
                __builtin_prefetch(&w[(size_t)ni[j] * K_EMB + lane], 0, 0);
            }
        }
        // 8 independent coalesced gathers + FMAs (in-flight together).
#pragma unroll
        for (int j = 0; j < K_CH; ++j) {
            acc = fmaf(cv[j], w[(size_t)ci[j] * K_EMB + lane], acc);
        }
        if (more) {
#pragma unroll
            for (int j = 0; j < K_CH; ++j) { ci[j] = ni[j]; cv[j] = nv[j]; }
        }
    }

    out[(size_t)b * K_EMB + lane] = acc + bias[lane];
}

extern "C" void kernel_launch(void* const* d_in, const int* in_sizes, int n_in,
                              void* d_out, int out_size, void* d_ws, size_t ws_size,
                              hipStream_t stream) {
    (void)n_in; (void)out_size; (void)d_ws; (void)ws_size;
    const int*   indices = (const int*)d_in[0];     // [B, NNZ] int32
    const float* values  = (const float*)d_in[1];   // [B, NNZ] f32
    const float* w       = (const float*)d_in[2];   // [VOCAB, EMB] f32
    const float* bias    = (const float*)d_in[3];   // [EMB] f32
    float*       out     = (float*)d_out;           // [B, EMB] f32

    const int B = in_sizes[0] / K_NNZ;              // 16384
    playlist_embedding_kernel<<<dim3(B), dim3(32), 0, stream>>>(
        indices, values, w, bias, out);
}